// RandomAffineCropSquareGPU_76536317215071
// MI455X (gfx1250) — compile-verified
//
#include <hip/hip_runtime.h>
#include <cstdint>
#include <cstddef>

#define TPB 256

namespace {

constexpr int W  = 512;   // input width
constexpr int H  = 512;   // input height
constexpr int S  = 224;   // output size
constexpr int T  = 32;    // output tile side per block (224 = 7*32)
constexpr int FP = 73;    // input footprint side: (T-1)*W/S = 70.86 -> 73 incl. bilinear +1
constexpr int LW = 73;    // LDS row stride (floats)
constexpr int CH = 3;
constexpr float PI_F = 3.14159265358979323846f;

// The corner (uniform) and per-pixel maps MUST be the same inlined expression
// so that floor-monotonicity makes every tap land inside [gx0, gx0+FP-1].
__device__ __forceinline__ float map_ix(float c, float s, float bx, float by) {
  return ((c * bx - s * by + 1.0f) * 512.0f - 1.0f) * 0.5f;
}
__device__ __forceinline__ float map_iy(float c, float s, float bx, float by) {
  return ((s * bx + c * by + 1.0f) * 512.0f - 1.0f) * 0.5f;
}
__device__ __forceinline__ float bxof(int p) {
  return ((float)p + 0.5f) * (2.0f / 224.0f) - 1.0f;
}
__device__ __forceinline__ int imin(int a, int b) { return a < b ? a : b; }
__device__ __forceinline__ int imax(int a, int b) { return a > b ? a : b; }

} // namespace

__global__ __launch_bounds__(TPB)
void affine_crop_async_kernel(const float* __restrict__ x,
                              const float* __restrict__ angles,
                              float* __restrict__ out)
{
  // 3 * 73 * 73 floats footprint + 4-float dummy sink for (never-read) OOB lanes
  __shared__ float tile[CH * FP * LW + 4];

  const int b   = blockIdx.z;
  const int tx0 = blockIdx.x * T;
  const int ty0 = blockIdx.y * T;

  const float rad = angles[b] * (PI_F / 180.0f);
  float sn, cs;
  sincosf(rad, &sn, &cs);                    // one shared argument reduction
  const float inv = 1.0f / (fabsf(cs) + fabsf(sn));
  const float c   = cs * inv;
  const float s   = sn * inv;

  // Uniform footprint origin: evaluate the affine map at the minimizing tile corner.
  const float ixmin = map_ix(c, s,
                             bxof(c >= 0.0f ? tx0 : tx0 + T - 1),
                             bxof(s >= 0.0f ? ty0 + T - 1 : ty0));
  const float iymin = map_iy(c, s,
                             bxof(s >= 0.0f ? tx0 : tx0 + T - 1),
                             bxof(c >= 0.0f ? ty0 : ty0 + T - 1));
  const int gx0 = (int)floorf(ixmin);
  const int gy0 = (int)floorf(iymin);

  // Zero-fill only needed when the footprint touches the image border
  // (zeroed cells reproduce grid_sample's zero-padding mask exactly).
  const bool full = (gx0 >= 0) && (gy0 >= 0) && (gx0 + FP <= W) && (gy0 + FP <= H);
  if (!full) {
    for (int i = (int)threadIdx.x; i < CH * FP * LW + 4; i += TPB) tile[i] = 0.0f;
  }
  __syncthreads();

  // Flat (low-32-bit) LDS byte addresses: ISA maps flat->LDS via addr[31:0].
  const unsigned lbase = (unsigned)(uintptr_t)(void*)&tile[0];
  const unsigned dummy = lbase + (unsigned)(CH * FP * LW) * 4u;

  // Stage footprint: coalesced async global->LDS B32 copies (ASYNCcnt path).
  {
    const float* srcb = x + (size_t)b * CH * H * W;
#pragma unroll
    for (int ch = 0; ch < CH; ++ch) {
      const float*   src  = srcb + (size_t)ch * H * W;
      const unsigned lch  = lbase + (unsigned)(ch * FP * LW) * 4u;
      for (int e = (int)threadIdx.x; e < FP * LW; e += TPB) {
        const int ry = e / LW;
        const int rx = e - ry * LW;
        const int gx = gx0 + rx;
        const int gy = gy0 + ry;
        const bool ok = (gx >= 0) & (gx < W) & (gy >= 0) & (gy < H);
        const int gxc = imin(imax(gx, 0), W - 1);
        const int gyc = imin(imax(gy, 0), H - 1);
        const unsigned long long ga =
            (unsigned long long)(uintptr_t)(src + (size_t)gyc * W + gxc);
        const unsigned la = ok ? (lch + (unsigned)e * 4u) : dummy;
        asm volatile("global_load_async_to_lds_b32 %0, %1, off"
                     :: "v"(la), "v"(ga) : "memory");
      }
    }
  }
  asm volatile("s_wait_asynccnt 0" ::: "memory");
  __syncthreads();

  // Bilinear sample from LDS; fully coalesced output stores.
  for (int k = 0; k < (T * T) / TPB; ++k) {
    const int p  = (int)threadIdx.x + k * TPB;
    const int lx = p & (T - 1);
    const int ly = p >> 5;
    const int px = tx0 + lx;
    const int py = ty0 + ly;

    const float bx = bxof(px);
    const float by = bxof(py);
    const float ix = map_ix(c, s, bx, by);
    const float iy = map_iy(c, s, bx, by);

    const float x0f = floorf(ix);
    const float y0f = floorf(iy);
    const float wx1 = ix - x0f, wx0 = 1.0f - wx1;
    const float wy1 = iy - y0f, wy0 = 1.0f - wy1;

    // Defensive clamp; mathematically already in [0, FP-2].
    const int lx0 = imin(imax((int)x0f - gx0, 0), FP - 2);
    const int ly0 = imin(imax((int)y0f - gy0, 0), FP - 2);
    const int tb  = ly0 * LW + lx0;

#pragma unroll
    for (int ch = 0; ch < CH; ++ch) {
      const float* tc = &tile[ch * (FP * LW) + tb];
      const float v00 = tc[0];
      const float v01 = tc[1];
      const float v10 = tc[LW];
      const float v11 = tc[LW + 1];
      const float res = (v00 * wx0 + v01 * wx1) * wy0 +
                        (v10 * wx0 + v11 * wx1) * wy1;
      out[(((size_t)b * CH + ch) * S + py) * S + px] = res;
    }
  }
}

extern "C" void kernel_launch(void* const* d_in, const int* in_sizes, int n_in,
                              void* d_out, int out_size, void* d_ws, size_t ws_size,
                              hipStream_t stream) {
  (void)in_sizes; (void)n_in; (void)out_size; (void)d_ws; (void)ws_size;
  const float* x      = (const float*)d_in[0];
  const float* angles = (const float*)d_in[1];
  float* out          = (float*)d_out;

  dim3 grid(S / T, S / T, 64);  // 7 x 7 tiles x 64 batches
  affine_crop_async_kernel<<<grid, TPB, 0, stream>>>(x, angles, out);
}